// GlobalSLC_44650480010013
// MI455X (gfx1250) — compile-verified
//
#include <hip/hip_runtime.h>

// Problem constants (from reference): B*T=48, N=512, C=64
#define N_NODES 512
#define C_DIM   64
#define BT      48
#define NC      (N_NODES * C_DIM)      // 32768 floats per (b,t) signal
#define NN      (N_NODES * N_NODES)    // 262144 floats per (b,t) Laplacian
#define LDS_STRIDE 68                  // padded row stride for transposed B tiles

typedef float v2f __attribute__((ext_vector_type(2)));
typedef float v8f __attribute__((ext_vector_type(8)));

// ---------------------------------------------------------------------------
// WMMA f32 16x16x4 wrapper (CDNA5: D = A(16x4) * B(4x16) + C(16x16), f32)
// ---------------------------------------------------------------------------
__device__ __forceinline__ v8f wmma4(v2f a, v2f b, v8f c) {
  return __builtin_amdgcn_wmma_f32_16x16x4_f32(
      /*neg_a=*/false, a, /*neg_b=*/false, b,
      /*c_mod=*/(short)0, c, /*reuse_a=*/false, /*reuse_b=*/false);
}

// A fragment: 16x4 tile of row-major A (row stride lda), rows mbase.., cols k0..
// ISA layout: lanes 0-15 -> M=lane, K=k0+{0,1}; lanes 16-31 -> M=lane-16, K=k0+{2,3}
__device__ __forceinline__ v2f load_a(const float* __restrict__ A, int lda,
                                      int mbase, int k0, int lane) {
  int m = mbase + (lane & 15);
  int k = k0 + ((lane >> 4) << 1);
  return *(const v2f*)(A + (size_t)m * lda + k);   // k even -> 8B aligned
}

// B fragment where B = X^T, X row-major (row stride ldx): B[k][n] = X[n][k]
__device__ __forceinline__ v2f load_bt(const float* __restrict__ X, int ldx,
                                       int k0, int nbase, int lane) {
  int n = nbase + (lane & 15);
  int k = k0 + ((lane >> 4) << 1);
  return *(const v2f*)(X + (size_t)n * ldx + k);
}

// B fragment from LDS-staged transposed tile sB[n][kk] (row stride LDS_STRIDE):
// element (kk, n) of the 64-row chunk. Single aligned 8-byte LDS read.
__device__ __forceinline__ v2f load_b_lds(const float* sB, int kk, int nbase, int lane) {
  int n  = nbase + (lane & 15);
  int kb = kk + ((lane >> 4) << 1);
  return *(const v2f*)(sB + (size_t)n * LDS_STRIDE + kb);
}

// ---------------------------------------------------------------------------
// Ordered-uint mapping so unsigned atomicMin/Max implement float min/max
// ---------------------------------------------------------------------------
__device__ __forceinline__ unsigned ford(float f) {
  unsigned u = __float_as_uint(f);
  return (u & 0x80000000u) ? ~u : (u | 0x80000000u);
}
__device__ __forceinline__ float funord(unsigned u) {
  return __uint_as_float((u & 0x80000000u) ? (u & 0x7FFFFFFFu) : ~u);
}
__device__ __forceinline__ unsigned umin2(unsigned a, unsigned b) { return a < b ? a : b; }
__device__ __forceinline__ unsigned umax2(unsigned a, unsigned b) { return a > b ? a : b; }

// ---------------------------------------------------------------------------
// Kernels
// ---------------------------------------------------------------------------
__global__ void k_init(unsigned* __restrict__ mm) {
  mm[0] = 0xFFFFFFFFu;  // ordered min identity
  mm[1] = 0x00000000u;  // ordered max identity
}

// out (+)= coef * (S @ W); S: (BT,512,64), W: (64,64).
// Block = 8 waves; each wave owns a 16x64 output strip (4 WMMA accumulators).
// W is staged once into LDS, transposed, shared by all waves.
__global__ void k_combine(const float* __restrict__ S, const float* __restrict__ W,
                          float coef, int accumulate, float* __restrict__ Dst) {
  __shared__ float sW[64 * LDS_STRIDE];
  int lane = threadIdx.x & 31;
  int wave = threadIdx.x >> 5;
  int bt   = blockIdx.x >> 2;                       // 4 blocks per (b,t)
  int tm   = (((blockIdx.x & 3) << 3) + wave) << 4; // strip row base, 0..496
  const float* Sb = S + (size_t)bt * NC;
  float* Db = Dst + (size_t)bt * NC;

  // Stage W (64x64) transposed: sW[n][k] = W[k][n]
#pragma unroll
  for (int r = 0; r < 64; r += 16) {
    int kk = (threadIdx.x >> 4) + r;
    int n4 = (threadIdx.x & 15) << 2;
    float4 v = *(const float4*)(W + (size_t)kk * C_DIM + n4);
    sW[(n4 + 0) * LDS_STRIDE + kk] = v.x;
    sW[(n4 + 1) * LDS_STRIDE + kk] = v.y;
    sW[(n4 + 2) * LDS_STRIDE + kk] = v.z;
    sW[(n4 + 3) * LDS_STRIDE + kk] = v.w;
  }
  __syncthreads();

  v8f acc[4] = {};
#pragma unroll 4
  for (int kk = 0; kk < C_DIM; kk += 4) {
    v2f a = load_a(Sb, C_DIM, tm, kk, lane);
#pragma unroll
    for (int c = 0; c < 4; ++c) {
      v2f b = load_b_lds(sW, kk, c << 4, lane);
      acc[c] = wmma4(a, b, acc[c]);
    }
  }

  int rowbase = tm + ((lane >> 4) << 3);            // lanes 16-31 hold rows r+8
#pragma unroll
  for (int c = 0; c < 4; ++c) {
    int n = (c << 4) + (lane & 15);
#pragma unroll
    for (int r = 0; r < 8; ++r) {
      float v = coef * acc[c][r];
      if (accumulate) v += Db[(size_t)(rowbase + r) * C_DIM + n];
      Db[(size_t)(rowbase + r) * C_DIM + n] = v;
    }
  }
}

// wd = u @ x^T per (b,t) + global min/max reduction of raw wd.
// One 16x16 tile per wave (both fragments are aligned v2f global loads).
__global__ void k_wd(const float* __restrict__ U, const float* __restrict__ X,
                     float* __restrict__ WD, unsigned* __restrict__ mm) {
  __shared__ unsigned smin[256];
  __shared__ unsigned smax[256];
  int lane = threadIdx.x & 31;
  int tile = blockIdx.x * 8 + (threadIdx.x >> 5);
  int bt = tile >> 10;                // 1024 tiles per (b,t): 32 x 32
  int t  = tile & 1023;
  int tm = (t >> 5) << 4;
  int tn = (t & 31) << 4;
  const float* A  = U + (size_t)bt * NC;
  const float* Xb = X + (size_t)bt * NC;
  float* D = WD + (size_t)bt * NN;
  v8f c = {};
#pragma unroll
  for (int k0 = 0; k0 < C_DIM; k0 += 4) {
    v2f a = load_a(A, C_DIM, tm, k0, lane);
    v2f b = load_bt(Xb, C_DIM, k0, tn, lane);  // B = x^T
    c = wmma4(a, b, c);
  }
  int n  = tn + (lane & 15);
  int mb = tm + ((lane >> 4) << 3);
  float lmin = c[0], lmax = c[0];
#pragma unroll
  for (int r = 0; r < 8; ++r) {
    D[(size_t)(mb + r) * N_NODES + n] = c[r];
    lmin = fminf(lmin, c[r]);
    lmax = fmaxf(lmax, c[r]);
  }
  smin[threadIdx.x] = ford(lmin);
  smax[threadIdx.x] = ford(lmax);
  __syncthreads();
  for (int s = 128; s > 0; s >>= 1) {
    if ((int)threadIdx.x < s) {
      smin[threadIdx.x] = umin2(smin[threadIdx.x], smin[threadIdx.x + s]);
      smax[threadIdx.x] = umax2(smax[threadIdx.x], smax[threadIdx.x + s]);
    }
    __syncthreads();
  }
  if (threadIdx.x == 0) {
    atomicMin(&mm[0], smin[0]);
    atomicMax(&mm[1], smax[0]);
  }
}

// sc[0] = a (global min), sc[1] = 1 / ((max + a) * N^2)
__global__ void k_prep(const unsigned* __restrict__ mm, float* __restrict__ sc) {
  float a  = funord(mm[0]);
  float mx = funord(mm[1]);
  sc[0] = a;
  sc[1] = 1.0f / ((mx + a) * 262144.0f);  // N^2 = 512*512
}

// wd = (wd + a) * inv, vectorized elementwise pass
__global__ void k_norm(float* __restrict__ WD, const float* __restrict__ sc) {
  float a = sc[0], inv = sc[1];
  size_t total  = (size_t)BT * NN;
  size_t stride = (size_t)gridDim.x * blockDim.x * 4;
  for (size_t i = ((size_t)blockIdx.x * blockDim.x + threadIdx.x) * 4; i < total; i += stride) {
    float4 v = *(float4*)(WD + i);
    v.x = (v.x + a) * inv;
    v.y = (v.y + a) * inv;
    v.z = (v.z + a) * inv;
    v.w = (v.w + a) * inv;
    *(float4*)(WD + i) = v;
  }
}

// D = alpha * (A @ B) + beta * Csub. A: 512x512 (per-bt stride strideA, 0 = shared ws),
// B/Csub/D: (BT,512,64). Block = 8 waves, one (b,t) slice of 8 x 16-row strips.
// B is staged into LDS in transposed 64-row chunks shared by all 8 waves;
// each wave keeps 4 accumulators (16x64 strip), reusing each A fragment 4x.
__global__ void k_chain(const float* __restrict__ A, size_t strideA,
                        const float* __restrict__ B, const float* __restrict__ Csub,
                        float alpha, float beta, float* __restrict__ D) {
  __shared__ float sB[64 * LDS_STRIDE];
  int lane = threadIdx.x & 31;
  int wave = threadIdx.x >> 5;
  int bt   = blockIdx.x >> 2;                       // 4 blocks per (b,t)
  int tm   = (((blockIdx.x & 3) << 3) + wave) << 4;
  const float* Ab = A + (size_t)bt * strideA;
  const float* Bb = B + (size_t)bt * NC;
  float* Db = D + (size_t)bt * NC;

  v8f acc[4] = {};
  for (int k0 = 0; k0 < N_NODES; k0 += 64) {
    __syncthreads();
    // Stage chunk: sB[n][kk] = B[k0+kk][n], kk in [0,64)
#pragma unroll
    for (int r = 0; r < 64; r += 16) {
      int kk = (threadIdx.x >> 4) + r;
      int n4 = (threadIdx.x & 15) << 2;
      float4 v = *(const float4*)(Bb + (size_t)(k0 + kk) * C_DIM + n4);
      sB[(n4 + 0) * LDS_STRIDE + kk] = v.x;
      sB[(n4 + 1) * LDS_STRIDE + kk] = v.y;
      sB[(n4 + 2) * LDS_STRIDE + kk] = v.z;
      sB[(n4 + 3) * LDS_STRIDE + kk] = v.w;
    }
    __syncthreads();
#pragma unroll 4
    for (int kk = 0; kk < 64; kk += 4) {
      v2f a = load_a(Ab, N_NODES, tm, k0 + kk, lane);
#pragma unroll
      for (int c = 0; c < 4; ++c) {
        v2f b = load_b_lds(sB, kk, c << 4, lane);
        acc[c] = wmma4(a, b, acc[c]);
      }
    }
  }

  int rowbase = tm + ((lane >> 4) << 3);
  const float* Cb = Csub ? (Csub + (size_t)bt * NC) : nullptr;
#pragma unroll
  for (int c = 0; c < 4; ++c) {
    int n = (c << 4) + (lane & 15);
#pragma unroll
    for (int r = 0; r < 8; ++r) {
      float v = alpha * acc[c][r];
      if (Cb) v += beta * Cb[(size_t)(rowbase + r) * C_DIM + n];
      Db[(size_t)(rowbase + r) * C_DIM + n] = v;
    }
  }
}

// ---------------------------------------------------------------------------
// Launcher
// ---------------------------------------------------------------------------
extern "C" void kernel_launch(void* const* d_in, const int* in_sizes, int n_in,
                              void* d_out, int out_size, void* d_ws, size_t ws_size,
                              hipStream_t stream) {
  (void)in_sizes; (void)n_in; (void)out_size; (void)ws_size;
  const float* x  = (const float*)d_in[0];  // (4,12,512,64)
  const float* ws = (const float*)d_in[1];  // (512,512)
  const float* ts = (const float*)d_in[2];  // (3,64,64)
  const float* wp = (const float*)d_in[3];  // (64,64)
  const float* td = (const float*)d_in[4];  // (6,64,64)
  float* out = (float*)d_out;               // (4,12,512,64)

  // Scratch layout
  unsigned* mm = (unsigned*)d_ws;           // [0]=ordered min, [1]=ordered max
  float* sc = (float*)d_ws + 2;             // a, inv
  float* wd = (float*)d_ws + 64;            // 48 * 512 * 512
  float* u  = wd + (size_t)BT * NN;         // 48 * 512 * 64 each below
  float* s1 = u  + (size_t)BT * NC;
  float* s2 = s1 + (size_t)BT * NC;
  float* v1 = s2 + (size_t)BT * NC;
  float* zA = v1 + (size_t)BT * NC;
  float* zB = zA + (size_t)BT * NC;

  const int STRIP_BLOCKS = BT * 4;          // 192: 8 strips/block, 32 strips per (b,t)
  const int WD_BLOCKS    = BT * 1024 / 8;   // 6144

  hipLaunchKernelGGL(k_init, dim3(1), dim3(1), 0, stream, mm);

  // u = x @ wp
  hipLaunchKernelGGL(k_combine, dim3(STRIP_BLOCKS), dim3(256), 0, stream, x, wp, 1.0f, 0, u);
  // wd_raw = u @ x^T, plus global min/max
  hipLaunchKernelGGL(k_wd, dim3(WD_BLOCKS), dim3(256), 0, stream, u, x, wd, mm);
  // scalars: a = min, inv = 1/((max+a)*N^2)
  hipLaunchKernelGGL(k_prep, dim3(1), dim3(1), 0, stream, mm, sc);
  // wd = (wd_raw + a) * inv
  hipLaunchKernelGGL(k_norm, dim3(2048), dim3(256), 0, stream, wd, sc);

  // Static branch signals: s1 = ws@x ; s2 = ws@s1  (tk2@x = 2*s2 - x)
  hipLaunchKernelGGL(k_chain, dim3(STRIP_BLOCKS), dim3(256), 0, stream,
                     ws, (size_t)0, x, (const float*)nullptr, 1.0f, 0.0f, s1);
  hipLaunchKernelGGL(k_chain, dim3(STRIP_BLOCKS), dim3(256), 0, stream,
                     ws, (size_t)0, s1, (const float*)nullptr, 1.0f, 0.0f, s2);
  // Dynamic branch signals: v1 = wd@x ; z2 = 2*wd@v1 - x
  hipLaunchKernelGGL(k_chain, dim3(STRIP_BLOCKS), dim3(256), 0, stream,
                     wd, (size_t)NN, x, (const float*)nullptr, 1.0f, 0.0f, v1);
  hipLaunchKernelGGL(k_chain, dim3(STRIP_BLOCKS), dim3(256), 0, stream,
                     wd, (size_t)NN, v1, x, 2.0f, -1.0f, zA);      // z2

  // out = x@ts0 + x@td0 - x@ts2 + s1@ts1 + 2*s2@ts2 + v1@td1 + z2@td2
  hipLaunchKernelGGL(k_combine, dim3(STRIP_BLOCKS), dim3(256), 0, stream, x,  ts + 0 * 4096,  1.0f, 0, out);
  hipLaunchKernelGGL(k_combine, dim3(STRIP_BLOCKS), dim3(256), 0, stream, x,  td + 0 * 4096,  1.0f, 1, out);
  hipLaunchKernelGGL(k_combine, dim3(STRIP_BLOCKS), dim3(256), 0, stream, x,  ts + 2 * 4096, -1.0f, 1, out);
  hipLaunchKernelGGL(k_combine, dim3(STRIP_BLOCKS), dim3(256), 0, stream, s1, ts + 1 * 4096,  1.0f, 1, out);
  hipLaunchKernelGGL(k_combine, dim3(STRIP_BLOCKS), dim3(256), 0, stream, s2, ts + 2 * 4096,  2.0f, 1, out);
  hipLaunchKernelGGL(k_combine, dim3(STRIP_BLOCKS), dim3(256), 0, stream, v1, td + 1 * 4096,  1.0f, 1, out);
  hipLaunchKernelGGL(k_combine, dim3(STRIP_BLOCKS), dim3(256), 0, stream, zA, td + 2 * 4096,  1.0f, 1, out);

  // z3 = 2*wd@z2 - v1 ; out += z3@td3
  hipLaunchKernelGGL(k_chain, dim3(STRIP_BLOCKS), dim3(256), 0, stream,
                     wd, (size_t)NN, zA, v1, 2.0f, -1.0f, zB);
  hipLaunchKernelGGL(k_combine, dim3(STRIP_BLOCKS), dim3(256), 0, stream, zB, td + 3 * 4096, 1.0f, 1, out);
  // z4 = 2*wd@z3 - v1 ; out += z4@td4
  hipLaunchKernelGGL(k_chain, dim3(STRIP_BLOCKS), dim3(256), 0, stream,
                     wd, (size_t)NN, zB, v1, 2.0f, -1.0f, zA);
  hipLaunchKernelGGL(k_combine, dim3(STRIP_BLOCKS), dim3(256), 0, stream, zA, td + 4 * 4096, 1.0f, 1, out);
  // z5 = 2*wd@z4 - v1 ; out += z5@td5
  hipLaunchKernelGGL(k_chain, dim3(STRIP_BLOCKS), dim3(256), 0, stream,
                     wd, (size_t)NN, zA, v1, 2.0f, -1.0f, zB);
  hipLaunchKernelGGL(k_combine, dim3(STRIP_BLOCKS), dim3(256), 0, stream, zB, td + 5 * 4096, 1.0f, 1, out);
}